// lnn_28750511080103
// MI455X (gfx1250) — compile-verified
//
#include <hip/hip_runtime.h>

// CDNA5 / gfx1250, wave32. Full-FP32 WMMA (V_WMMA_F32_16X16X4_F32) for the
// batched 64x64 (and 2->64) MLP layers of a Lagrangian-NN RK2 step.

typedef float v2f __attribute__((ext_vector_type(2)));
typedef float v4f __attribute__((ext_vector_type(4)));
typedef float v8f __attribute__((ext_vector_type(8)));

#define DT      0.05f
#define ALPHA   (2.0f/3.0f)
#define LOG2E   1.4426950408889634f
#define LN2     0.6931471805599453f

// ---------------- LDS layout (float indices) ----------------
// W2 matrices are stored in "B-tile pair" layout:
//   element W2[n][k]  ->  lds[(kk*64 + n)*2 + (k&1)],  kk = k>>1
// so a lane's B fragment for K-chunk kb (k = 4*kb .. 4*kb+3) is one
// contiguous float2 at (kk*64+n)*2 with kk = 2*kb + (lane>=16).
#define OFF_W2B_L 0
#define OFF_W2B_V 4096
#define OFF_W1_L  8192
#define OFF_W1_V  8320
#define OFF_B1_L  8448
#define OFF_B2_L  8512
#define OFF_B1_V  8576
#define OFF_B2_V  8640
#define OFF_W3_L  8704
#define OFF_W3_V  8768
#define OFF_B3_L  8832
#define OFF_ACT   8840
#define HS        68                 // padded activation row stride (floats)
#define ACT_PER_WAVE (2*16*HS)       // h-buffer + g-buffer per wave
#define WAVES 8
#define SMEM_FLOATS (OFF_ACT + WAVES*ACT_PER_WAVE)   // 26248 floats ~ 102.5 KB

// D = A(16x4) * B(4x16) + C, all fp32.
// Fragment layouts (per CDNA5 ISA §7.12.2):
//   A(m,k): lane = m + 16*(k>=2), vgpr = k&1
//   B(k,n): lane = n + 16*(k>=2), vgpr = k&1
//   C/D(m,n): vgpr = m&7, lane = n + 16*(m>=8)
static __device__ __forceinline__ v8f wmma4(v2f a, v2f b, v8f c) {
  return __builtin_amdgcn_wmma_f32_16x16x4_f32(false, a, false, b, (short)0, c,
                                               false, false);
}

// softplus + sigmoid sharing one v_exp_f32 / v_log_f32 / v_rcp_f32, and one
// sign predicate (v_cmp reused by both selects; avoids fmax canonicalize).
//   e = 2^(-|z|*log2e) = exp(-|z|)   (in [0,1], never denormal; 1+e in [1,2])
//   softplus(z) = relu(z) + ln2*log2(1+e)
//   sigmoid(z)  = (z>=0 ? 1 : e) * rcp(1+e)
static __device__ __forceinline__ void sp_sig(float z, float& sp, float& sig) {
  bool pos = z >= 0.f;
  float e  = __builtin_amdgcn_exp2f(-fabsf(z) * LOG2E);
  float r  = __builtin_amdgcn_rcpf(1.f + e);
  sig = (pos ? 1.f : e) * r;
  sp  = (pos ? z : 0.f) + LN2 * __builtin_amdgcn_logf(1.f + e);
}

// One derivs() evaluation: returns acceleration for this lane's element.
// Whole wave cooperates (WMMA + LDS transposes); lanes 16..31 mirror 0..15.
// MUST be force-inlined so InferAddressSpaces can prove all sm/hbuf/gbuf
// accesses are LDS (ds_* ops, not flat_*).
static __device__ __forceinline__ float
lnn_acc(float* sm, float* hbuf, float* gbuf,
        float q, float qdot, float aext, int lh, int hi) {
  float cq = __cosf(q), sq = __sinf(q);
  float lm = hi ? 0.f : 1.f;     // lanes 16..31 feed the zero K=2,3 padding
  float Lv = 0.f, dLv = 0.f, dVdq = 0.f;

  for (int net = 0; net < 2; ++net) {
    const float* W2B = sm + (net ? OFF_W2B_V : OFF_W2B_L);
    const float* W1  = sm + (net ? OFF_W1_V  : OFF_W1_L);
    const float* B1  = sm + (net ? OFF_B1_V  : OFF_B1_L);
    const float* B2  = sm + (net ? OFF_B2_V  : OFF_B2_L);
    const float* W3  = sm + (net ? OFF_W3_V  : OFF_W3_L);

    // ---- layer 1: [cos q, sin q] -> 64, fwd + tangent (dt/dq = [-sin, cos])
    v2f aF = {  cq * lm, sq * lm };
    v2f aT = { -sq * lm, cq * lm };
#pragma unroll
    for (int t = 0; t < 4; ++t) {
      int n = t*16 + lh;
      v2f w1 = *(const v2f*)(W1 + 2*n);          // unconditional ds_load_b64
      v2f b  = { w1.x * lm, w1.y * lm };
      float bias = B1[n];
      v8f cb, cz;
#pragma unroll
      for (int r = 0; r < 8; ++r) { cb[r] = bias; cz[r] = 0.f; }
      v8f zf = wmma4(aF, b, cb);     // z1  = W1 t + b1
      v8f zt = wmma4(aT, b, cz);     // dz1 = W1 dt
#pragma unroll
      for (int r = 0; r < 8; ++r) {
        int m = r + 8*hi;
        float sp, sg;
        sp_sig(zf[r], sp, sg);
        hbuf[m*HS + n] = sp;
        gbuf[m*HS + n] = sg * zt[r];
      }
    }

    // ---- layer 2: 64 -> 64 GEMM, fwd + tangent, K in chunks of 4
    v8f accF[4], accT[4];
#pragma unroll
    for (int t = 0; t < 4; ++t) {
      float bias = B2[t*16 + lh];
#pragma unroll
      for (int r = 0; r < 8; ++r) { accF[t][r] = bias; accT[t][r] = 0.f; }
    }
    for (int kb = 0; kb < 16; ++kb) {
      int kbase = kb*4;
      int kk    = kb*2 + hi;
      v2f aF2 = *(const v2f*)(hbuf + lh*HS + kbase + 2*hi);
      v2f aT2 = *(const v2f*)(gbuf + lh*HS + kbase + 2*hi);
#pragma unroll
      for (int t = 0; t < 4; ++t) {
        int n = t*16 + lh;
        v2f b = *(const v2f*)(W2B + (kk*64 + n)*2);
        accF[t] = wmma4(aF2, b, accF[t]);
        accT[t] = wmma4(aT2, b, accT[t]);
      }
    }
#pragma unroll
    for (int t = 0; t < 4; ++t) {
      int n = t*16 + lh;
#pragma unroll
      for (int r = 0; r < 8; ++r) {
        int m = r + 8*hi;
        float sp, sg;
        sp_sig(accF[t][r], sp, sg);
        hbuf[m*HS + n] = sp;
        gbuf[m*HS + n] = sg * accT[t][r];
      }
    }

    // ---- layer 3: 64 -> 1 dot products, split across half-waves:
    // lanes 0..15 accumulate the forward dot (hbuf), lanes 16..31 the
    // tangent dot (gbuf); one __shfl_xor(16) swap hands both halves both
    // results (all lanes stay correct mirrors).
    const float* dbuf = hi ? gbuf : hbuf;
    float v = 0.f;
#pragma unroll
    for (int k4 = 0; k4 < 16; ++k4) {
      v4f wv = *(const v4f*)(W3 + k4*4);
      v4f xv = *(const v4f*)(dbuf + lh*HS + k4*4);
      v = fmaf(wv.x, xv.x, v); v = fmaf(wv.y, xv.y, v);
      v = fmaf(wv.z, xv.z, v); v = fmaf(wv.w, xv.w, v);
    }
    float other = __shfl_xor(v, 16, 32);
    float y  = hi ? other : v;   // forward-pass output
    float dy = hi ? v : other;   // tangent output (d/dq)
    if (net == 0) { Lv = y + sm[OFF_B3_L]; dLv = dy; }
    else          { dVdq = dy; }
  }

  // N=1, OUT_L=1: M = L^2,  c = L*dL*qdot^2,  acc = (a - c - dV/dq)/M
  float c = Lv * dLv * qdot * qdot;
  return (aext - c - dVdq) / (Lv * Lv);
}

__global__ void __launch_bounds__(WAVES*32)
lnn_rk2_kernel(const float* __restrict__ o,    const float* __restrict__ a,
               const float* __restrict__ W1_L, const float* __restrict__ b1_L,
               const float* __restrict__ W2_L, const float* __restrict__ b2_L,
               const float* __restrict__ W3_L, const float* __restrict__ b3_L,
               const float* __restrict__ W1_V, const float* __restrict__ b1_V,
               const float* __restrict__ W2_V, const float* __restrict__ b2_V,
               const float* __restrict__ W3_V, float* __restrict__ out) {
  extern __shared__ float sm[];
  const int tid = threadIdx.x;

  // ---- cooperative weight staging into LDS ----
  for (int idx = tid; idx < 4096; idx += WAVES*32) {
    int n = idx >> 6, k = idx & 63;
    int kk = k >> 1, j = k & 1;
    sm[OFF_W2B_L + (kk*64 + n)*2 + j] = W2_L[idx];
    sm[OFF_W2B_V + (kk*64 + n)*2 + j] = W2_V[idx];
  }
  for (int idx = tid; idx < 128; idx += WAVES*32) {
    sm[OFF_W1_L + idx] = W1_L[idx];
    sm[OFF_W1_V + idx] = W1_V[idx];
  }
  for (int idx = tid; idx < 64; idx += WAVES*32) {
    sm[OFF_B1_L + idx] = b1_L[idx];
    sm[OFF_B2_L + idx] = b2_L[idx];
    sm[OFF_B1_V + idx] = b1_V[idx];
    sm[OFF_B2_V + idx] = b2_V[idx];
    sm[OFF_W3_L + idx] = W3_L[idx];
    sm[OFF_W3_V + idx] = W3_V[idx];
  }
  if (tid == 0) sm[OFF_B3_L] = b3_L[0];
  __syncthreads();

  const int lane = tid & 31, lh = lane & 15, hi = lane >> 4, wave = tid >> 5;
  float* hbuf = sm + OFF_ACT + wave*ACT_PER_WAVE;
  float* gbuf = hbuf + 16*HS;

  const int g = blockIdx.x*(WAVES*16) + wave*16 + lh;  // lanes 16..31 mirror
  float o0   = o[3*g + 0];
  float o1   = o[3*g + 1];
  float qdot = o[3*g + 2];
  float aext = a[g];
  float q    = atan2f(o1, o0);

  // RK2 (alpha = 2/3): s1 = s + dt*(0.25*k1 + 0.75*k2)
  float k1  = lnn_acc(sm, hbuf, gbuf, q, qdot, aext, lh, hi);
  float q2  = q    + (ALPHA*DT)*qdot;
  float qd2 = qdot + (ALPHA*DT)*k1;
  float k2  = lnn_acc(sm, hbuf, gbuf, q2, qd2, aext, lh, hi);

  float s1q  = q    + DT*(0.25f*qdot + 0.75f*qd2);
  float s1qd = qdot + DT*(0.25f*k1   + 0.75f*k2);

  if (lane < 16) {
    out[3*g + 0] = __cosf(s1q);
    out[3*g + 1] = __sinf(s1q);
    out[3*g + 2] = s1qd;
  }
}

extern "C" void kernel_launch(void* const* d_in, const int* in_sizes, int n_in,
                              void* d_out, int out_size, void* d_ws,
                              size_t ws_size, hipStream_t stream) {
  (void)n_in; (void)out_size; (void)d_ws; (void)ws_size;
  const float* o    = (const float*)d_in[0];
  const float* a    = (const float*)d_in[1];
  const float* W1_L = (const float*)d_in[2];
  const float* b1_L = (const float*)d_in[3];
  const float* W2_L = (const float*)d_in[4];
  const float* b2_L = (const float*)d_in[5];
  const float* W3_L = (const float*)d_in[6];
  const float* b3_L = (const float*)d_in[7];
  const float* W1_V = (const float*)d_in[8];
  const float* b1_V = (const float*)d_in[9];
  const float* W2_V = (const float*)d_in[10];
  const float* b2_V = (const float*)d_in[11];
  const float* W3_V = (const float*)d_in[12];
  float* out = (float*)d_out;

  const int B = in_sizes[1];                  // a: (B,1)
  dim3 grid((unsigned)(B / (WAVES*16)));      // 16 elems per wave
  dim3 block(WAVES*32);
  size_t smem = (size_t)SMEM_FLOATS * sizeof(float);
  lnn_rk2_kernel<<<grid, block, smem, stream>>>(o, a, W1_L, b1_L, W2_L, b2_L,
                                                W3_L, b3_L, W1_V, b1_V, W2_V,
                                                b2_V, W3_V, out);
}